// GraphSAGE_9646496547063
// MI455X (gfx1250) — compile-verified
//
#include <hip/hip_runtime.h>
#include <hip/hip_bf16.h>

typedef __attribute__((ext_vector_type(2))) float v2f;
typedef __attribute__((ext_vector_type(8))) float v8f;
typedef int v4i __attribute__((vector_size(16)));   // int32x4, matches builtin param

#if defined(__has_builtin)
#if __has_builtin(__builtin_amdgcn_global_load_async_to_lds_b128)
#define HAVE_ASYNC_LDS 1
#endif
#endif

// ---------------------------------------------------------------------------
// GEMM: Y[row, COLOFF+col] = act( X[M,K] @ W[K,NW] + bias[NW] )
// All strides are template constants so the fully-unrolled K-loop folds every
// address into global_load immediate offsets (no per-iteration u64 math).
// One wave computes a (16*RT) x 16 output strip: the B 4x16 fragment is
// reused across RT row tiles -> 4 WMMAs per 6 loads per K-step.
// f32 16x16x4 fragment layout (ISA 7.12.2): lanes 0-15 hold K={0,1},
// lanes 16-31 hold K={2,3}; A: M = lane&15, B: N = lane&15.
// Column guard (NW=40 head) uses clamped address + value select, never exec.
// ---------------------------------------------------------------------------
template<int K, int NW, int LDY, int COLOFF, int RELU, int RT>
__global__ __launch_bounds__(32) void wmma_gemm(
    const float* __restrict__ X, const float* __restrict__ W,
    const float* __restrict__ bias, float* __restrict__ Y)
{
    const int lane = threadIdx.x & 31;
    const int half = lane >> 4;             // 0 or 1
    const int l15  = lane & 15;
    const int row0 = blockIdx.y * (16 * RT);
    const int col0 = blockIdx.x * 16;

    const int kb   = half << 1;             // K sub-offset: 0 or 2
    const int col  = col0 + l15;
    const bool cv  = (col < NW);
    const int colc = cv ? col : 0;          // clamped, always in-bounds

    const float* wp = W + (size_t)kb * NW + colc;      // B base for this lane
    const float* xp[RT];
    v8f acc[RT];
    #pragma unroll
    for (int r = 0; r < RT; ++r) {
        acc[r] = (v8f){};
        xp[r] = X + (size_t)(row0 + r * 16 + l15) * K + kb;  // A base per row tile
    }

    #pragma unroll
    for (int k0 = 0; k0 < K; k0 += 4) {
        v2f b;                               // rows k0+kb, k0+kb+1 of W
        b.x = wp[(size_t)k0 * NW];           // compile-time immediate offsets
        b.y = wp[(size_t)(k0 + 1) * NW];
        if (!cv) { b.x = 0.0f; b.y = 0.0f; } // v_cndmask, not exec branch
        #pragma unroll
        for (int r = 0; r < RT; ++r) {
            v2f a = *(const v2f*)(xp[r] + k0);   // 8B vector load, imm offset
            acc[r] = __builtin_amdgcn_wmma_f32_16x16x4_f32(
                false, a, false, b, (short)0, acc[r], false, false);
        }
    }

    if (cv) {
        const float bv = bias[col];
        #pragma unroll
        for (int r = 0; r < RT; ++r) {
            float* yp = Y + (size_t)(row0 + r * 16 + (half << 3)) * LDY + COLOFF + col;
            #pragma unroll
            for (int i = 0; i < 8; ++i) {
                float v = acc[r][i] + bv;
                if (RELU) v = fmaxf(v, 0.0f);
                yp[(size_t)i * LDY] = v;     // imm-offset stores
            }
        }
    }
}

// ---------------------------------------------------------------------------
// agg[v,c] = max over u with A[v,u] of P[u,c]; isolated nodes -> 0.
// One block per node v; adjacency row staged into LDS — via the gfx1250
// async global->LDS path (GLOBAL_LOAD_ASYNC_TO_LDS_B128, ASYNCcnt) when the
// builtin exists, else a plain copy loop. word==0 gives a wave-uniform skip.
// C must equal blockDim.x (128).
// ---------------------------------------------------------------------------
__global__ __launch_bounds__(128) void maskmax_kernel(
    const float* __restrict__ P, const unsigned char* __restrict__ A,
    float* __restrict__ agg, int Nn, int C)
{
    __shared__ unsigned int arow[256];   // 1024 adjacency bytes
    const int v   = blockIdx.x;
    const int tid = threadIdx.x;
    const int words = Nn >> 2;           // 256

#ifdef HAVE_ASYNC_LDS
    if (tid < (Nn >> 4)) {               // 64 lanes x 16B = 1024B row
        v4i* gp = (v4i*)(uintptr_t)(const void*)(A + (size_t)v * Nn + tid * 16);
        __attribute__((address_space(3))) v4i* lp =
            (__attribute__((address_space(3))) v4i*)(unsigned)(uintptr_t)(const void*)(&arow[tid * 4]);
        __builtin_amdgcn_global_load_async_to_lds_b128(gp, lp, 0, 0);
    }
#if __has_builtin(__builtin_amdgcn_s_wait_asynccnt)
    __builtin_amdgcn_s_wait_asynccnt(0);
#else
    asm volatile("s_wait_asynccnt 0" ::: "memory");
#endif
    __syncthreads();
#else
    const unsigned int* arow_g = (const unsigned int*)(A + (size_t)v * Nn);
    for (int i = tid; i < words; i += blockDim.x) arow[i] = arow_g[i];
    __syncthreads();
#endif

    const int c = tid;                   // one channel per thread
    float mx = -1e30f;
    for (int w = 0; w < words; ++w) {
        const unsigned int mw = arow[w];
        if (mw == 0u) continue;          // uniform across the wave
        const int ub = w << 2;
        if (mw & 0x000000FFu) mx = fmaxf(mx, P[(size_t)(ub + 0) * C + c]);
        if (mw & 0x0000FF00u) mx = fmaxf(mx, P[(size_t)(ub + 1) * C + c]);
        if (mw & 0x00FF0000u) mx = fmaxf(mx, P[(size_t)(ub + 2) * C + c]);
        if (mw & 0xFF000000u) mx = fmaxf(mx, P[(size_t)(ub + 3) * C + c]);
    }
    if (mx <= -5e29f) mx = 0.0f;         // isolated node
    agg[(size_t)v * C + c] = mx;
}

// ---------------------------------------------------------------------------
// In-place row L2 normalization: H[row,:] /= max(||H[row,:]||, 1e-12).
// One wave per row (C=128 -> one float4 per lane), wave32 shuffle reduction.
// ---------------------------------------------------------------------------
__global__ __launch_bounds__(256) void rownorm_kernel(
    float* __restrict__ Hm, int Mrows, int C)
{
    const int wave = threadIdx.x >> 5;
    const int lane = threadIdx.x & 31;
    const int row  = blockIdx.x * 8 + wave;
    if (row >= Mrows) return;

    float4* p = (float4*)(Hm + (size_t)row * C) + lane;
    float4 v = *p;
    float s = v.x * v.x + v.y * v.y + v.z * v.z + v.w * v.w;
    #pragma unroll
    for (int off = 16; off > 0; off >>= 1) s += __shfl_xor(s, off, 32);
    const float inv = 1.0f / fmaxf(sqrtf(s), 1e-12f);
    v.x *= inv; v.y *= inv; v.z *= inv; v.w *= inv;
    *p = v;
}

// ---------------------------------------------------------------------------
// Orchestration. Inputs (setup_inputs order):
// 0:x[1024,128] f32  1:A[1024,1024] bool(1B)  2..7: Wp1,bp1,Ws1,bs1,Wn1,bn1
// 8..13: Wp2,bp2,Ws2,bs2,Wn2,bn2  14:Wh  15:bh.  Output: [1024,40] f32.
// ---------------------------------------------------------------------------
extern "C" void kernel_launch(void* const* d_in, const int* in_sizes, int n_in,
                              void* d_out, int out_size, void* d_ws, size_t ws_size,
                              hipStream_t stream)
{
    const float*         x   = (const float*)d_in[0];
    const unsigned char* A   = (const unsigned char*)d_in[1];  // jax bool = 1 byte
    const float* Wp1 = (const float*)d_in[2];
    const float* bp1 = (const float*)d_in[3];
    const float* Ws1 = (const float*)d_in[4];
    const float* bs1 = (const float*)d_in[5];
    const float* Wn1 = (const float*)d_in[6];
    const float* bn1 = (const float*)d_in[7];
    const float* Wp2 = (const float*)d_in[8];
    const float* bp2 = (const float*)d_in[9];
    const float* Ws2 = (const float*)d_in[10];
    const float* bs2 = (const float*)d_in[11];
    const float* Wn2 = (const float*)d_in[12];
    const float* bn2 = (const float*)d_in[13];
    const float* Wh  = (const float*)d_in[14];
    const float* bh  = (const float*)d_in[15];
    float* out = (float*)d_out;

    constexpr int Nn = 1024, DIN = 128, H = 64, D2 = 128, NCLS = 40, RT = 4;

    float* ws     = (float*)d_ws;
    float* pooled = ws;                    // 1024*128
    float* agg    = ws + 131072;           // 1024*128
    float* h1     = ws + 262144;           // 1024*128
    float* h2     = ws + 393216;           // 1024*128

    const dim3 blk32(32), blk128(128), blk256(256);
    auto g = [&](int nw) { return dim3((unsigned)((nw + 15) / 16), (unsigned)(Nn / (16 * RT))); };

    // ---- layer 1 ----
    wmma_gemm<DIN, DIN, DIN, 0, 1, RT><<<g(DIN), blk32, 0, stream>>>(x, Wp1, bp1, pooled);
    maskmax_kernel<<<dim3(Nn), blk128, 0, stream>>>(pooled, A, agg, Nn, DIN);
    wmma_gemm<DIN, H, D2, 0, 1, RT><<<g(H), blk32, 0, stream>>>(x,   Ws1, bs1, h1);
    wmma_gemm<DIN, H, D2, H, 1, RT><<<g(H), blk32, 0, stream>>>(agg, Wn1, bn1, h1);
    rownorm_kernel<<<dim3(Nn / 8), blk256, 0, stream>>>(h1, Nn, D2);

    // ---- layer 2 ----
    wmma_gemm<D2, D2, D2, 0, 1, RT><<<g(D2), blk32, 0, stream>>>(h1, Wp2, bp2, pooled);
    maskmax_kernel<<<dim3(Nn), blk128, 0, stream>>>(pooled, A, agg, Nn, D2);
    wmma_gemm<D2, H, D2, 0, 1, RT><<<g(H), blk32, 0, stream>>>(h1,  Ws2, bs2, h2);
    wmma_gemm<D2, H, D2, H, 1, RT><<<g(H), blk32, 0, stream>>>(agg, Wn2, bn2, h2);
    rownorm_kernel<<<dim3(Nn / 8), blk256, 0, stream>>>(h2, Nn, D2);

    // ---- classification head (N=40, guarded tiles) ----
    wmma_gemm<D2, NCLS, NCLS, 0, 0, RT><<<g(NCLS), blk32, 0, stream>>>(h2, Wh, bh, out);
}